// GNN_Multi_Layer_15882789060739
// MI455X (gfx1250) — compile-verified
//
#include <hip/hip_runtime.h>

typedef float v2f __attribute__((ext_vector_type(2)));
typedef float v8f __attribute__((ext_vector_type(8)));

#define C_IN  40
#define C_HID 64
#define C_OUT 40
#define LD2   48   // padded N for layer-2 GEMM output

static __device__ __forceinline__ void atomAddF(float* p, float v) {
    unsafeAtomicAdd(p, v);   // non-returning global_atomic_add_f32
}

__global__ void fill_f32(float* __restrict__ p, float v, long long n) {
    long long i  = (long long)blockIdx.x * blockDim.x + threadIdx.x;
    long long st = (long long)gridDim.x * blockDim.x;
    for (; i < n; i += st) p[i] = v;
}

__global__ void deg_accum(const long long* __restrict__ dst, const float* __restrict__ ew,
                          float* __restrict__ deg, int E) {
    int i  = blockIdx.x * blockDim.x + threadIdx.x;
    int st = gridDim.x * blockDim.x;
    for (; i < E; i += st) atomAddF(&deg[(int)dst[i]], ew[i]);
}

__global__ void deg_to_dinv(float* __restrict__ deg, int n) {
    int i  = blockIdx.x * blockDim.x + threadIdx.x;
    int st = gridDim.x * blockDim.x;
    for (; i < n; i += st) {
        float d = deg[i];
        deg[i] = d > 0.f ? rsqrtf(d) : 0.f;
    }
}

__global__ void edge_norm(const long long* __restrict__ src, const long long* __restrict__ dst,
                          const float* __restrict__ ew, const float* __restrict__ dinv,
                          float* __restrict__ norm, int E) {
    int i  = blockIdx.x * blockDim.x + threadIdx.x;
    int st = gridDim.x * blockDim.x;
    for (; i < E; i += st)
        norm[i] = dinv[(int)src[i]] * ew[i] * dinv[(int)dst[i]];
}

// ----- dense GEMM via V_WMMA_F32_16X16X4_F32 --------------------------------
// out[M x NPAD] = A[M x K] * W[K x NOUT], W zero-padded to NPAD cols in LDS.
// blockDim = 256 (8 waves); each wave computes a 16 x NPAD tile.
template<int K, int NOUT, int NPAD>
__global__ void gemm_wmma(const float* __restrict__ A, const float* __restrict__ W,
                          float* __restrict__ out, int M) {
    __shared__ float sW[K * NPAD];
    for (int idx = threadIdx.x; idx < K * NPAD; idx += blockDim.x) {
        int r = idx / NPAD, c = idx % NPAD;
        sW[idx] = (c < NOUT) ? W[r * NOUT + c] : 0.f;
    }
    __syncthreads();

    constexpr int NT = NPAD / 16;
    const int wave  = threadIdx.x >> 5;
    const int lane  = threadIdx.x & 31;
    const int half  = lane >> 4;     // 0: lanes 0-15, 1: lanes 16-31
    const int l16   = lane & 15;
    const int m0    = blockIdx.x * 128 + wave * 16;
    const int mrow  = m0 + l16;
    const bool valid = (mrow < M);
    const float* arow = A + (size_t)(valid ? mrow : (M - 1)) * K;

    v8f acc[NT];
#pragma unroll
    for (int t = 0; t < NT; ++t)
        acc[t] = (v8f){0.f, 0.f, 0.f, 0.f, 0.f, 0.f, 0.f, 0.f};

#pragma unroll
    for (int k4 = 0; k4 < K; k4 += 4) {
        // A fragment: lane L -> A[m0 + L%16][k4 + 2*(L/16) + {0,1}]
        v2f av = *(const v2f*)(arow + k4 + half * 2);
        v2f a;
        a.x = valid ? av.x : 0.f;
        a.y = valid ? av.y : 0.f;
#pragma unroll
        for (int t = 0; t < NT; ++t) {
            // B fragment: lane L -> W[k4 + 2*(L/16) + {0,1}][t*16 + L%16]
            v2f b;
            b.x = sW[(k4 + half * 2 + 0) * NPAD + t * 16 + l16];
            b.y = sW[(k4 + half * 2 + 1) * NPAD + t * 16 + l16];
            acc[t] = __builtin_amdgcn_wmma_f32_16x16x4_f32(
                false, a, false, b, (short)0, acc[t], false, false);
        }
    }

    // D layout: lane L, VGPR v -> row m0 + v + 8*(L/16), col t*16 + L%16
#pragma unroll
    for (int t = 0; t < NT; ++t) {
#pragma unroll
        for (int v = 0; v < 8; ++v) {
            int row = m0 + v + 8 * half;
            if (row < M) out[(size_t)row * NPAD + t * 16 + l16] = acc[t][v];
        }
    }
}

// ----- edge scatter: acc[dst] += norm[e] * xin[src] -------------------------
template<int C, int LDI, int LDO>
__global__ void scatter_edges(const long long* __restrict__ src, const long long* __restrict__ dst,
                              const float* __restrict__ norm, const float* __restrict__ xin,
                              float* __restrict__ acc, long long total) {
    long long i  = (long long)blockIdx.x * blockDim.x + threadIdx.x;
    long long st = (long long)gridDim.x * blockDim.x;
    for (; i < total; i += st) {
        int e = (int)(i / C);
        int f = (int)(i % C);
        int s = (int)src[e];
        int d = (int)dst[e];
        atomAddF(&acc[(size_t)d * LDO + f], norm[e] * xin[(size_t)s * LDI + f]);
    }
}

// h = relu(acc + dinv^2 * xl + b), in place into acc (C_HID features, ld 64)
__global__ void epilogue_relu(float* __restrict__ acc, const float* __restrict__ xl,
                              const float* __restrict__ dinv, const float* __restrict__ bias,
                              int n) {
    long long total = (long long)n * C_HID;
    long long i  = (long long)blockIdx.x * blockDim.x + threadIdx.x;
    long long st = (long long)gridDim.x * blockDim.x;
    for (; i < total; i += st) {
        int node = (int)(i >> 6);
        int f    = (int)(i & 63);
        float di = dinv[node];
        float v  = acc[i] + di * di * xl[i] + bias[f];
        acc[i] = v > 0.f ? v : 0.f;
    }
}

// out += dinv^2 * xl2 + b2, in place (out ld 40, xl2 ld 48)
__global__ void epilogue_out(float* __restrict__ out, const float* __restrict__ xl2,
                             const float* __restrict__ dinv, const float* __restrict__ bias,
                             int n) {
    long long total = (long long)n * C_OUT;
    long long i  = (long long)blockIdx.x * blockDim.x + threadIdx.x;
    long long st = (long long)gridDim.x * blockDim.x;
    for (; i < total; i += st) {
        int node = (int)(i / C_OUT);
        int f    = (int)(i % C_OUT);
        float di = dinv[node];
        out[i] = out[i] + di * di * xl2[(size_t)node * LD2 + f] + bias[f];
    }
}

static inline size_t align256(size_t x) { return (x + 255) & ~(size_t)255; }

extern "C" void kernel_launch(void* const* d_in, const int* in_sizes, int n_in,
                              void* d_out, int out_size, void* d_ws, size_t ws_size,
                              hipStream_t stream) {
    const float*     x    = (const float*)d_in[0];
    const long long* eidx = (const long long*)d_in[1];   // int64 in reference
    const float*     ew   = (const float*)d_in[2];
    const float*     W1   = (const float*)d_in[3];
    const float*     b1   = (const float*)d_in[4];
    const float*     W2   = (const float*)d_in[5];
    const float*     b2   = (const float*)d_in[6];
    float*           out  = (float*)d_out;

    const int n = in_sizes[0] / C_IN;     // 100000
    const int E = in_sizes[2];            // 1600000
    const long long* srcp = eidx;
    const long long* dstp = eidx + E;

    char* ws = (char*)d_ws;
    size_t off = 0;
    float* dinv = (float*)(ws + off); off += align256((size_t)n * 4);
    float* norm = (float*)(ws + off); off += align256((size_t)E * 4);
    float* xl   = (float*)(ws + off); off += align256((size_t)n * C_HID * 4);
    float* acc  = (float*)(ws + off); off += align256((size_t)n * C_HID * 4);

    const int TB = 256;

    // degree (init 1.0 for self-loop) -> dinv -> per-edge norm
    fill_f32<<<1024, TB, 0, stream>>>(dinv, 1.0f, (long long)n);
    deg_accum<<<2048, TB, 0, stream>>>(dstp, ew, dinv, E);
    deg_to_dinv<<<1024, TB, 0, stream>>>(dinv, n);
    edge_norm<<<2048, TB, 0, stream>>>(srcp, dstp, ew, dinv, norm, E);

    const int gemm_blocks = (n + 127) / 128;

    // ---- layer 1: xl = x @ W1; acc = scatter(norm * xl[src]); h = relu(...)
    gemm_wmma<C_IN, C_HID, C_HID><<<gemm_blocks, TB, 0, stream>>>(x, W1, xl, n);
    hipMemsetAsync(acc, 0, (size_t)n * C_HID * 4, stream);
    scatter_edges<C_HID, C_HID, C_HID><<<4096, TB, 0, stream>>>(
        srcp, dstp, norm, xl, acc, (long long)E * C_HID);
    epilogue_relu<<<4096, TB, 0, stream>>>(acc, xl, dinv, b1, n);   // acc now holds h

    // ---- layer 2: xl2 = h @ W2 (reuse xl buffer, ld 48); scatter into d_out
    gemm_wmma<C_HID, C_OUT, LD2><<<gemm_blocks, TB, 0, stream>>>(acc, W2, xl, n);
    hipMemsetAsync(out, 0, (size_t)n * C_OUT * 4, stream);
    scatter_edges<C_OUT, LD2, C_OUT><<<4096, TB, 0, stream>>>(
        srcp, dstp, norm, xl, out, (long long)E * C_OUT);
    epilogue_out<<<4096, TB, 0, stream>>>(out, xl, dinv, b2, n);
}